// SinglePopEncoder_77000173683381
// MI455X (gfx1250) — compile-verified
//
#include <hip/hip_runtime.h>

// SinglePopEncoder spike kernel for MI455X (gfx1250).
// Pure streaming elementwise op: ~812 MB moved, ~0.5 GFLOP -> HBM-bound
// (~35us at 23.3 TB/s). No contraction dim exists, so WMMA is inapplicable;
// we optimize the data path instead: b128 non-temporal streams + CDNA5
// async global->LDS staging of the tiny W/b tables.

typedef __attribute__((ext_vector_type(4))) float v4f;
typedef __attribute__((ext_vector_type(4))) int   v4i;

typedef __attribute__((address_space(1))) v4i* g_v4i_ptr;  // global (AS1)
typedef __attribute__((address_space(3))) v4i* l_v4i_ptr;  // LDS    (AS3)

#define TPB  96     // 3 wave32s; 96 float4 lanes == one 384-float output row
#define ROWS 64     // rows (batch elements) per block

__global__ __launch_bounds__(TPB) void snn_spike_encode(
    const float* __restrict__ x,     // (B, 6)
    const float* __restrict__ W,     // (6, 64)
    const float* __restrict__ bias,  // (6, 64)
    const float* __restrict__ mem,   // (B, 6, 64)
    float* __restrict__ out,         // (B, 384)
    int B)
{
    __shared__ float sW[384];
    __shared__ float sB[384];
    const int t = threadIdx.x;  // 0..95

    // ---- Stage W and b (1536 B each) into LDS via the gfx1250 async engine.
    // Each of the 96 lanes moves one 16 B chunk of W and one of b.
#if __has_builtin(__builtin_amdgcn_global_load_async_to_lds_b128)
    __builtin_amdgcn_global_load_async_to_lds_b128(
        (g_v4i_ptr)(W + t * 4),
        (l_v4i_ptr)(sW + t * 4), 0, 0);
    __builtin_amdgcn_global_load_async_to_lds_b128(
        (g_v4i_ptr)(bias + t * 4),
        (l_v4i_ptr)(sB + t * 4), 0, 0);
  #if __has_builtin(__builtin_amdgcn_s_wait_asynccnt)
    __builtin_amdgcn_s_wait_asynccnt(0);
  #else
    asm volatile("s_wait_asynccnt 0x0" ::: "memory");
  #endif
#else
    *(v4f*)(sW + t * 4) = *(const v4f*)(W + t * 4);
    *(v4f*)(sB + t * 4) = *(const v4f*)(bias + t * 4);
#endif
    __syncthreads();

    // Fixed per-thread channel/population slot: c = t/16, p = 4*(t%16).
    const int   c    = t >> 4;
    const float beta = (c & 1) ? 0.85f : 0.9f;   // BETA = {.9,.85,.9,.85,.9,.85}
    const v4f   W4   = *(const v4f*)(sW + t * 4);
    const v4f   b4   = *(const v4f*)(sB + t * 4);

    const long long row0 = (long long)blockIdx.x * ROWS;

#pragma unroll 4
    for (int i = 0; i < ROWS; ++i) {
        const long long r = row0 + i;
        if (r >= B) break;

        const float     xv  = x[r * 6 + c];          // 16 lanes broadcast-share
        const long long off = r * 384 + t * 4;

        // Streaming read of mem: non-temporal (TH=NT), b128.
        v4f m = __builtin_nontemporal_load((const v4f*)(mem + off));

        v4f o;
#pragma unroll
        for (int k = 0; k < 4; ++k) {
            const float mm    = m[k];
            const float cur   = __builtin_fmaf(xv, W4[k], b4[k]);  // x*W + b
            const float reset = (mm > 1.0f) ? 1.0f : 0.0f;         // thr = 1
            const float mnew  = __builtin_fmaf(beta, mm, cur) - reset;
            o[k] = (mnew > 1.0f) ? 1.0f : 0.0f;                    // heaviside
        }

        // Streaming write of spikes: non-temporal, b128.
        __builtin_nontemporal_store(o, (v4f*)(out + off));
    }
}

extern "C" void kernel_launch(void* const* d_in, const int* in_sizes, int n_in,
                              void* d_out, int out_size, void* d_ws, size_t ws_size,
                              hipStream_t stream)
{
    (void)n_in; (void)out_size; (void)d_ws; (void)ws_size;

    const float* x    = (const float*)d_in[0];   // (B, 6)
    const float* W    = (const float*)d_in[1];   // (6, 64)
    const float* bias = (const float*)d_in[2];   // (6, 64)
    const float* mem  = (const float*)d_in[3];   // (B, 6, 64)
    float*       out  = (float*)d_out;           // (B, 384)

    const int B    = in_sizes[0] / 6;            // 262144
    const int grid = (B + ROWS - 1) / ROWS;      // 4096 blocks of 96 threads

    hipLaunchKernelGGL(snn_spike_encode, dim3(grid), dim3(TPB), 0, stream,
                       x, W, bias, mem, out, B);
}